// SinkhornDistance_uniform_89833535963741
// MI455X (gfx1250) — compile-verified
//
#include <hip/hip_runtime.h>
#include <hip/hip_bf16.h>
#include <math.h>

typedef __attribute__((ext_vector_type(2))) float v2f;
typedef __attribute__((ext_vector_type(8))) float v8f;

#define BB   8
#define PP   1024
#define DD   256
#define EPSV 0.1f
#define INV_EPS 10.0f
#define LOG_MU  (-6.9314694f)   /* log(1/1024 + 1e-8) */
#define N_ITER  100
#define ERR_TOTAL_THRESH 0.8f   /* THRESH(0.1) * B(8): mean over batch */

// ---------------------------------------------------------------- norms
__global__ void norms_kernel(const float* __restrict__ x, float* __restrict__ nrm) {
    int row  = blockIdx.x * (blockDim.x >> 5) + (threadIdx.x >> 5);
    int lane = threadIdx.x & 31;
    if (row >= BB * PP) return;
    const float* p = x + (size_t)row * DD;
    float s = 0.f;
    for (int k = lane; k < DD; k += 32) { float v = p[k]; s += v * v; }
    for (int off = 16; off > 0; off >>= 1) s += __shfl_down(s, off, 32);
    if (lane == 0) nrm[row] = sqrtf(s);
}

// ------------------------------------------- cost matrix via f32 WMMA
__global__ void __launch_bounds__(32)
cost_wmma_kernel(const float* __restrict__ x, const float* __restrict__ y,
                 const float* __restrict__ nx, const float* __restrict__ ny,
                 float* __restrict__ Cmat) {
    __shared__ float sx[16 * 260];   // padded stride 260: bank-conflict free
    __shared__ float sy[16 * 260];
    const int b  = blockIdx.z;
    const int i0 = blockIdx.y * 16;
    const int j0 = blockIdx.x * 16;
    const int lane = threadIdx.x;

    const float4* gx = (const float4*)(x + ((size_t)b * PP + i0) * DD);
    const float4* gy = (const float4*)(y + ((size_t)b * PP + j0) * DD);
    for (int t = lane; t < 16 * 64; t += 32) {
        int r = t >> 6, c = t & 63;
        *((float4*)&sx[r * 260 + c * 4]) = gx[r * 64 + c];
        *((float4*)&sy[r * 260 + c * 4]) = gy[r * 64 + c];
    }
    __syncthreads();

    const int m  = lane & 15;          // A row / B column within tile
    const int kk = (lane >> 4) << 1;   // K sub-offset per half-wave
    const float* ax = &sx[m * 260 + kk];
    const float* by = &sy[m * 260 + kk];

    v8f acc = {};
    #pragma unroll 8
    for (int k = 0; k < DD; k += 4) {
        v2f a, bfr;
        a.x   = ax[k];  a.y   = ax[k + 1];
        bfr.x = by[k];  bfr.y = by[k + 1];
        acc = __builtin_amdgcn_wmma_f32_16x16x4_f32(
                  false, a, false, bfr, (short)0, acc, false, false);
    }

    // C/D layout: VGPR r holds M = r + 8*(lane>=16), N = lane&15
    const int nhalf = lane >> 4;
    const int col   = lane & 15;
    const float nyv = ny[b * PP + j0 + col];
    #pragma unroll
    for (int r = 0; r < 8; ++r) {
        int row = r + 8 * nhalf;
        float nxv   = nx[b * PP + i0 + row];
        float denom = fmaxf(nxv * nyv, 1e-8f);
        Cmat[((size_t)b * PP + i0 + row) * PP + (j0 + col)] = 1.0f - acc[r] / denom;
    }
}

// ------------------------------------------- Sinkhorn row update (u)
__global__ void row_update_kernel(const float* __restrict__ C,
                                  float* __restrict__ u,
                                  const float* __restrict__ v,
                                  float* __restrict__ errbuf,
                                  const int* __restrict__ done) {
    if (*done) return;
    __shared__ float red[256];
    const int b = blockIdx.y, i = blockIdx.x, tid = threadIdx.x;
    const float ui = u[b * PP + i];
    const float4 c4 = ((const float4*)(C + ((size_t)b * PP + i) * PP))[tid];
    const float4 v4 = ((const float4*)(v + b * PP))[tid];

    float m0 = (-c4.x + ui + v4.x) * INV_EPS;
    float m1 = (-c4.y + ui + v4.y) * INV_EPS;
    float m2 = (-c4.z + ui + v4.z) * INV_EPS;
    float m3 = (-c4.w + ui + v4.w) * INV_EPS;

    float mx = fmaxf(fmaxf(m0, m1), fmaxf(m2, m3));
    red[tid] = mx; __syncthreads();
    for (int s = 128; s > 0; s >>= 1) { if (tid < s) red[tid] = fmaxf(red[tid], red[tid + s]); __syncthreads(); }
    const float mxa = red[0]; __syncthreads();

    float sm = expf(m0 - mxa) + expf(m1 - mxa) + expf(m2 - mxa) + expf(m3 - mxa);
    red[tid] = sm; __syncthreads();
    for (int s = 128; s > 0; s >>= 1) { if (tid < s) red[tid] += red[tid + s]; __syncthreads(); }

    if (tid == 0) {
        float lse = logf(red[0]) + mxa;
        float un  = EPSV * (LOG_MU - lse) + ui;
        errbuf[b * PP + i] = fabsf(un - ui);
        u[b * PP + i] = un;
    }
}

// ------------------------------------------- Sinkhorn col update (v)
__global__ void col_update_kernel(const float* __restrict__ C,
                                  const float* __restrict__ u,
                                  float* __restrict__ v,
                                  const int* __restrict__ done) {
    if (*done) return;
    __shared__ float sm[8][32], ss[8][32];
    const int b  = blockIdx.y;
    const int tx = threadIdx.x, ty = threadIdx.y;
    const int j  = blockIdx.x * 32 + tx;
    const float vj = v[b * PP + j];

    float rm = -INFINITY, rs = 0.f;
    for (int r = 0; r < 128; ++r) {
        int i = ty * 128 + r;
        float Mv = (-C[((size_t)b * PP + i) * PP + j] + u[b * PP + i] + vj) * INV_EPS;
        float nm = fmaxf(rm, Mv);
        rs = rs * expf(rm - nm) + expf(Mv - nm);
        rm = nm;
    }
    sm[ty][tx] = rm; ss[ty][tx] = rs;
    __syncthreads();
    if (ty == 0) {
        float m = sm[0][tx], s = ss[0][tx];
        #pragma unroll
        for (int w = 1; w < 8; ++w) {
            float m2 = sm[w][tx], s2 = ss[w][tx];
            float nm = fmaxf(m, m2);
            s = s * expf(m - nm) + s2 * expf(m2 - nm);
            m = nm;
        }
        float lse = logf(s) + m;
        v[b * PP + j] = EPSV * (LOG_MU - lse) + vj;   // nu == mu (uniform, same P)
    }
}

// ------------------------------------------- convergence check (deterministic)
__global__ void check_done_kernel(const float* __restrict__ errbuf, int* done) {
    if (*done) return;
    __shared__ float red[256];
    const int tid = threadIdx.x;
    float s = 0.f;
    for (int k = tid; k < BB * PP; k += 256) s += errbuf[k];
    red[tid] = s; __syncthreads();
    for (int st = 128; st > 0; st >>= 1) { if (tid < st) red[tid] += red[tid + st]; __syncthreads(); }
    if (tid == 0 && red[0] < ERR_TOTAL_THRESH) *done = 1;
}

// ------------------------------------------- finalize: pi (in place) + partial cost
__global__ void finalize_kernel(float* __restrict__ PI,
                                const float* __restrict__ u,
                                const float* __restrict__ v,
                                float* __restrict__ partial) {
    __shared__ float red[256];
    const int b   = blockIdx.y;
    const int tid = threadIdx.x;
    const size_t base = (size_t)b * PP * PP;
    const int t = blockIdx.x * 256 + tid;   // float4 index
    const int e = t * 4;
    const int i = e >> 10;
    const int jq = (e & 1023) >> 2;

    float4 c4 = ((const float4*)(PI + base))[t];
    float4 v4 = ((const float4*)(v + b * PP))[jq];
    const float ui = u[b * PP + i];
    float4 p;
    p.x = expf((-c4.x + ui + v4.x) * INV_EPS);
    p.y = expf((-c4.y + ui + v4.y) * INV_EPS);
    p.z = expf((-c4.z + ui + v4.z) * INV_EPS);
    p.w = expf((-c4.w + ui + v4.w) * INV_EPS);
    ((float4*)(PI + base))[t] = p;

    red[tid] = p.x * c4.x + p.y * c4.y + p.z * c4.z + p.w * c4.w;
    __syncthreads();
    for (int st = 128; st > 0; st >>= 1) { if (tid < st) red[tid] += red[tid + st]; __syncthreads(); }
    if (tid == 0) partial[b * 1024 + blockIdx.x] = red[0];
}

__global__ void cost_reduce_kernel(const float* __restrict__ partial, float* __restrict__ cost) {
    __shared__ float red[256];
    const int b = blockIdx.x, tid = threadIdx.x;
    float s = 0.f;
    for (int k = tid; k < 1024; k += 256) s += partial[b * 1024 + k];
    red[tid] = s; __syncthreads();
    for (int st = 128; st > 0; st >>= 1) { if (tid < st) red[tid] += red[tid + st]; __syncthreads(); }
    if (tid == 0) cost[b] = red[0];
}

__global__ void init_kernel(float* u, float* v, int* done) {
    int t = threadIdx.x;
    for (int k = t; k < BB * PP; k += 256) { u[k] = 0.f; v[k] = 0.f; }
    if (t == 0) *done = 0;
}

// ---------------------------------------------------------------- launch
extern "C" void kernel_launch(void* const* d_in, const int* in_sizes, int n_in,
                              void* d_out, int out_size, void* d_ws, size_t ws_size,
                              hipStream_t stream) {
    const float* x = (const float*)d_in[0];
    const float* y = (const float*)d_in[1];
    float* out  = (float*)d_out;
    float* cost = out;              // 8 floats
    float* Cpi  = out + 8;          // 8*1024*1024 floats: C, then overwritten by pi

    float* ws      = (float*)d_ws;
    float* nx      = ws;            // 8192
    float* ny      = ws + 8192;     // 8192
    float* u       = ws + 16384;    // 8192
    float* v       = ws + 24576;    // 8192
    float* errbuf  = ws + 32768;    // 8192
    float* partial = ws + 40960;    // 8192
    int*   done    = (int*)(ws + 49152);

    init_kernel<<<1, 256, 0, stream>>>(u, v, done);
    norms_kernel<<<1024, 256, 0, stream>>>(x, nx);
    norms_kernel<<<1024, 256, 0, stream>>>(y, ny);
    cost_wmma_kernel<<<dim3(PP / 16, PP / 16, BB), 32, 0, stream>>>(x, y, nx, ny, Cpi);

    for (int it = 0; it < N_ITER; ++it) {
        row_update_kernel<<<dim3(PP, BB), 256, 0, stream>>>(Cpi, u, v, errbuf, done);
        col_update_kernel<<<dim3(PP / 32, BB), dim3(32, 8), 0, stream>>>(Cpi, u, v, done);
        check_done_kernel<<<1, 256, 0, stream>>>(errbuf, done);
    }

    finalize_kernel<<<dim3(1024, BB), 256, 0, stream>>>(Cpi, u, v, partial);
    cost_reduce_kernel<<<BB, 256, 0, stream>>>(partial, cost);
}